// GNNEncoder_18021682774172
// MI455X (gfx1250) — compile-verified
//
#include <hip/hip_runtime.h>
#include <hip/hip_bf16.h>

// ---------------------------------------------------------------------------
// GCN encoder for MI455X (gfx1250, wave32).
//   layer: m = h @ W (WMMA f32 16x16x4) ; agg = scatter-add(norm * m[src] -> dst)
//          h' = act(agg + m*dis^2 + b)   (self-loop folded in analytically)
// Aggregation is L2-resident (feature matrices ~20.5MB << 192MB L2).
// ---------------------------------------------------------------------------

typedef __attribute__((ext_vector_type(2))) float v2f;
typedef __attribute__((ext_vector_type(8))) float v8f;

#define N_NODES 40000
#define N_EDGES 640000
#define IN_DIM  3
#define HID_DIM 128
#define LAT_DIM 64

// ---------------------------------------------------------------- zero fill
__global__ void zero_kernel(float* __restrict__ p, int n) {
    int i = blockIdx.x * blockDim.x + threadIdx.x;
    int stride = gridDim.x * blockDim.x;
    for (; i < n; i += stride) p[i] = 0.0f;
}

// ------------------------------------------------------- degree accumulation
__global__ void degree_kernel(const int* __restrict__ dst, float* __restrict__ deg,
                              int nEdges) {
    int e = blockIdx.x * blockDim.x + threadIdx.x;
    if (e < nEdges) atomicAdd(&deg[dst[e]], 1.0f);
}

// deg includes +1 self loop; deg >= 1 always, so no max() needed.
__global__ void rsqrt_kernel(float* __restrict__ dis, int n) {
    int i = blockIdx.x * blockDim.x + threadIdx.x;
    if (i < n) dis[i] = rsqrtf(dis[i] + 1.0f);
}

// --------------------------------------------------------------- WMMA GEMM
// C(MxN) = A(MxK, row-major, lda=K) * B(KxN, row-major).
// One wave per 16x16 C tile; K stepped by 4 via V_WMMA_F32_16X16X4_F32.
// M % 16 == 0 and N % 16 == 0 for all call sites -> EXEC all-ones at WMMA.
template <int K>
__global__ __launch_bounds__(256) void gemm_wmma_f32(
    const float* __restrict__ A, const float* __restrict__ B,
    float* __restrict__ C, int M, int N) {
    const int lane   = threadIdx.x & 31;
    const int wave   = threadIdx.x >> 5;
    const int tilesN = N >> 4;
    const int tile   = blockIdx.x * 8 + wave;
    const int totalTiles = (M >> 4) * tilesN;
    if (tile >= totalTiles) return;           // uniform per wave

    const int tm = tile / tilesN;
    const int tn = tile % tilesN;

    // A fragment lane mapping (ISA 7.12.2, 32-bit A 16x4):
    //   lanes 0-15 : M=lane,    VGPR0=K0, VGPR1=K1
    //   lanes 16-31: M=lane-16, VGPR0=K2, VGPR1=K3
    const int arow  = tm * 16 + (lane & 15);
    const int khalf = (lane >> 4) * 2;        // 0 or 2
    // B fragment (4x16): row striped across lanes within a VGPR (mirror of A).
    const int ncol  = tn * 16 + (lane & 15);

    v8f acc = {};
    for (int k0 = 0; k0 < K; k0 += 4) {
        const int ka = k0 + khalf;
        v2f a, b;
        a.x = (ka     < K) ? A[(size_t)arow * K + ka]     : 0.0f;
        a.y = (ka + 1 < K) ? A[(size_t)arow * K + ka + 1] : 0.0f;
        b.x = (ka     < K) ? B[(size_t)(ka)     * N + ncol] : 0.0f;
        b.y = (ka + 1 < K) ? B[(size_t)(ka + 1) * N + ncol] : 0.0f;
        // (neg_a, A, neg_b, B, c_mod, C, reuse_a, reuse_b)
        acc = __builtin_amdgcn_wmma_f32_16x16x4_f32(
            false, a, false, b, (short)0, acc, false, false);
    }

    // C/D layout: VGPR r -> row = 8*(lane>=16)+r, col = lane&15.
    const int crow = tm * 16 + (lane >> 4) * 8;
#pragma unroll
    for (int r = 0; r < 8; ++r)
        C[(size_t)(crow + r) * N + ncol] = acc[r];
}

// ---------------------------------------------------- edge scatter-aggregate
// One wave per edge: coalesced float4 gather of m[src], f32 atomics into agg[dst].
template <int DIM>
__global__ __launch_bounds__(256) void scatter_edges(
    const int* __restrict__ src, const int* __restrict__ dst,
    const float* __restrict__ dis, const float* __restrict__ m,
    float* __restrict__ agg, int nEdges) {
    const int gw   = (blockIdx.x * blockDim.x + threadIdx.x) >> 5;
    const int lane = threadIdx.x & 31;
    if (gw >= nEdges) return;

    const int s = src[gw];
    const int d = dst[gw];
    const float coeff = dis[s] * dis[d];

    constexpr int PER = DIM / 32;             // 4 (hid) or 2 (lat)
    const float* __restrict__ mrow = m + (size_t)s * DIM + lane * PER;
    float* __restrict__ arow = agg + (size_t)d * DIM + lane * PER;

    float v[PER];
#pragma unroll
    for (int j = 0; j < PER; ++j) v[j] = mrow[j];   // combines to b64/b128 load
#pragma unroll
    for (int j = 0; j < PER; ++j) atomicAdd(&arow[j], v[j] * coeff);
}

// ------------------------------------------- bias + self-loop + activation
template <int DIM, bool RELU>
__global__ void finalize_kernel(const float* __restrict__ agg,
                                const float* __restrict__ m,
                                const float* __restrict__ dis,
                                const float* __restrict__ bias,
                                float* __restrict__ out, int nNodes) {
    int idx = blockIdx.x * blockDim.x + threadIdx.x;
    if (idx >= nNodes * DIM) return;
    const int node = idx / DIM;
    const int d    = idx % DIM;
    const float di = dis[node];
    float v = agg[idx] + m[idx] * di * di + bias[d];   // self-loop folded in
    if (RELU) v = fmaxf(v, 0.0f);
    out[idx] = v;
}

// ---------------------------------------------------------------------------
extern "C" void kernel_launch(void* const* d_in, const int* in_sizes, int n_in,
                              void* d_out, int out_size, void* d_ws, size_t ws_size,
                              hipStream_t stream) {
    const float* x   = (const float*)d_in[0];
    const int*   ei  = (const int*)d_in[1];
    const float* W0  = (const float*)d_in[2];
    const float* b0  = (const float*)d_in[3];
    const float* W1  = (const float*)d_in[4];
    const float* b1  = (const float*)d_in[5];
    const float* W2  = (const float*)d_in[6];
    const float* b2  = (const float*)d_in[7];
    float* out = (float*)d_out;

    const int* src = ei;            // edge_index[0]
    const int* dst = ei + N_EDGES;  // edge_index[1]

    // workspace partition (floats): dis | m | agg | h  (~62 MB total)
    float* ws  = (float*)d_ws;
    float* dis = ws;
    float* m   = dis + N_NODES;
    float* agg = m   + (size_t)N_NODES * HID_DIM;
    float* h   = agg + (size_t)N_NODES * HID_DIM;

    const int T = 256;
    const int nodeDimBlocksH = (N_NODES * HID_DIM + T - 1) / T;
    const int nodeDimBlocksL = (N_NODES * LAT_DIM + T - 1) / T;
    const int edgeWaveBlocks = (N_EDGES * 32 + T - 1) / T;   // 1 wave / edge
    const int gemmBlocksH = (N_NODES / 16) * (HID_DIM / 16) / 8;  // 2500
    const int gemmBlocksL = (N_NODES / 16) * (LAT_DIM / 16) / 8;  // 1250

    // ---- normalization coefficients: dis = rsqrt(indeg + 1) ----
    zero_kernel<<<512, T, 0, stream>>>(dis, N_NODES);
    degree_kernel<<<(N_EDGES + T - 1) / T, T, 0, stream>>>(dst, dis, N_EDGES);
    rsqrt_kernel<<<(N_NODES + T - 1) / T, T, 0, stream>>>(dis, N_NODES);

    // ---- layer 0: x(40000x3) @ W0(3x128) -> relu ----
    gemm_wmma_f32<IN_DIM><<<gemmBlocksH, T, 0, stream>>>(x, W0, m, N_NODES, HID_DIM);
    zero_kernel<<<2048, T, 0, stream>>>(agg, N_NODES * HID_DIM);
    scatter_edges<HID_DIM><<<edgeWaveBlocks, T, 0, stream>>>(src, dst, dis, m, agg, N_EDGES);
    finalize_kernel<HID_DIM, true><<<nodeDimBlocksH, T, 0, stream>>>(agg, m, dis, b0, h, N_NODES);

    // ---- layer 1: h(40000x128) @ W1(128x128) -> relu ----
    gemm_wmma_f32<HID_DIM><<<gemmBlocksH, T, 0, stream>>>(h, W1, m, N_NODES, HID_DIM);
    zero_kernel<<<2048, T, 0, stream>>>(agg, N_NODES * HID_DIM);
    scatter_edges<HID_DIM><<<edgeWaveBlocks, T, 0, stream>>>(src, dst, dis, m, agg, N_EDGES);
    finalize_kernel<HID_DIM, true><<<nodeDimBlocksH, T, 0, stream>>>(agg, m, dis, b1, h, N_NODES);

    // ---- layer 2: h(40000x128) @ W2(128x64) -> out ----
    gemm_wmma_f32<HID_DIM><<<gemmBlocksL, T, 0, stream>>>(h, W2, m, N_NODES, LAT_DIM);
    zero_kernel<<<1024, T, 0, stream>>>(agg, N_NODES * LAT_DIM);
    scatter_edges<LAT_DIM><<<edgeWaveBlocks, T, 0, stream>>>(src, dst, dis, m, agg, N_EDGES);
    finalize_kernel<LAT_DIM, false><<<nodeDimBlocksL, T, 0, stream>>>(agg, m, dis, b2, out, N_NODES);
}